// Model_34823594836411
// MI455X (gfx1250) — compile-verified
//
#include <hip/hip_runtime.h>
#include <math.h>

#define TPB 256

// ---------------------------------------------------------------------------
// Pass 0: zero the per-node accumulator [n_nodes x float4] in d_ws.
// ---------------------------------------------------------------------------
__global__ void zero_acc_kernel(float* __restrict__ acc, int n) {
    int i = blockIdx.x * blockDim.x + threadIdx.x;
    if (i < n) acc[i] = 0.0f;
}

// ---------------------------------------------------------------------------
// Pass 1: per-edge scatter.
//   Stage edge_src/edge_dst index streams into LDS with the CDNA5 async
//   copy engine (global_load_async_to_lds_b32, tracked by ASYNCcnt), then
//   gather positions, compute the unit direction sh1 = rel/max(|rel|,eps),
//   and scatter-add {sh1.x, sh1.y, sh1.z, 1.0} into acc[dst] with hardware
//   global_atomic_add_f32 (all four land in one 16B-aligned cacheline).
//   w0/w1/node_feat factor out of the segment sum (s is gathered by dst,
//   hence constant per segment) and are applied in the finalize pass.
// ---------------------------------------------------------------------------
__global__ __launch_bounds__(TPB) void edge_scatter_kernel(
    const float* __restrict__ pos,
    const int*   __restrict__ esrc,
    const int*   __restrict__ edst,
    float*       __restrict__ acc,
    int n_edges)
{
    __shared__ int s_src[TPB];
    __shared__ int s_dst[TPB];

    const int tid  = threadIdx.x;
    const int idx  = blockIdx.x * TPB + tid;
    // Clamp so every lane issues a valid async copy (keeps EXEC uniform for
    // the staging; the atomic scatter below is predicated on idx validity).
    const int gidx = (idx < n_edges) ? idx : (n_edges - 1);

    // LDS byte offsets of this lane's staging slots. For a generic pointer to
    // LDS, addr[31:0] is the wave-relative LDS offset (CDNA5 ISA §10.2).
    unsigned lds_src = (unsigned)(size_t)&s_src[tid];
    unsigned lds_dst = (unsigned)(size_t)&s_dst[tid];
    unsigned voff    = (unsigned)gidx * 4u;   // byte offset from SGPR base (GVS mode)

    // Async global -> LDS copies (ASYNCcnt), one b32 per lane per stream.
    asm volatile("global_load_async_to_lds_b32 %0, %1, %2"
                 :: "v"(lds_src), "v"(voff), "s"(esrc) : "memory");
    asm volatile("global_load_async_to_lds_b32 %0, %1, %2"
                 :: "v"(lds_dst), "v"(voff), "s"(edst) : "memory");
    // Each lane consumes only its own slot -> per-wave async wait suffices.
    asm volatile("s_wait_asynccnt 0" ::: "memory");

    const int src = s_src[tid];
    const int dst = s_dst[tid];

    // Position gathers: 1.2 MB table is L2-resident on MI455X (192 MB L2).
    const float* ps = pos + 3 * (size_t)src;
    const float* pd = pos + 3 * (size_t)dst;
    const float rx = pd[0] - ps[0];
    const float ry = pd[1] - ps[1];
    const float rz = pd[2] - ps[2];

    const float n2  = fmaf(rx, rx, fmaf(ry, ry, rz * rz));
    const float inv = 1.0f / fmaxf(sqrtf(n2), 1e-12f);   // matches rel/max(|rel|,EPS)

    if (idx < n_edges) {
        float* a = acc + 4 * (size_t)dst;                // 16B-aligned float4 slot
        unsafeAtomicAdd(a + 0, rx * inv);                // global_atomic_add_f32
        unsafeAtomicAdd(a + 1, ry * inv);
        unsafeAtomicAdd(a + 2, rz * inv);
        unsafeAtomicAdd(a + 3, 1.0f);                    // edge count
    }
}

// ---------------------------------------------------------------------------
// Pass 2: per-node finalize.
//   c = count; invc = 1/max(c,1); s = node_feat[n]
//   out0 = w0 * s * c * invc            (sum of E identical w0*s, then mean;
//                                        == 0 when c == 0, matching reference)
//   out1 = w1 * s * (sum sh1) * invc
// ---------------------------------------------------------------------------
__global__ void finalize_kernel(const float4* __restrict__ acc,
                                const float*  __restrict__ feat,
                                const float*  __restrict__ w0,
                                const float*  __restrict__ w1,
                                float*        __restrict__ out,
                                int n_nodes)
{
    int n = blockIdx.x * blockDim.x + threadIdx.x;
    if (n >= n_nodes) return;

    const float4 a   = acc[n];
    const float  c    = a.w;
    const float  invc = 1.0f / fmaxf(c, 1.0f);
    const float  s    = feat[n];
    const float  sv   = s * invc;

    float* o = out + 4 * (size_t)n;
    o[0] = w0[0] * s * c * invc;
    o[1] = w1[0] * sv * a.x;
    o[2] = w1[1] * sv * a.y;
    o[3] = w1[2] * sv * a.z;
}

// ---------------------------------------------------------------------------
// Inputs (setup_inputs order):
//   0: positions [N,3] f32   1: node_feat [N,1] f32   2: w0 [1] f32
//   3: w1 [3] f32            4: edge_src [E] i32      5: edge_dst [E] i32
// Output: [N,4] f32.  Workspace: n_nodes * float4 accumulator.
// ---------------------------------------------------------------------------
extern "C" void kernel_launch(void* const* d_in, const int* in_sizes, int n_in,
                              void* d_out, int out_size, void* d_ws, size_t ws_size,
                              hipStream_t stream) {
    const float* pos  = (const float*)d_in[0];
    const float* feat = (const float*)d_in[1];
    const float* w0   = (const float*)d_in[2];
    const float* w1   = (const float*)d_in[3];
    const int*   esrc = (const int*)d_in[4];
    const int*   edst = (const int*)d_in[5];

    const int n_nodes = in_sizes[0] / 3;
    const int n_edges = in_sizes[4];

    float* acc = (float*)d_ws;            // n_nodes * 4 floats (1.6 MB)
    float* out = (float*)d_out;

    const int acc_elems = 4 * n_nodes;
    zero_acc_kernel<<<(acc_elems + 255) / 256, 256, 0, stream>>>(acc, acc_elems);

    edge_scatter_kernel<<<(n_edges + TPB - 1) / TPB, TPB, 0, stream>>>(
        pos, esrc, edst, acc, n_edges);

    finalize_kernel<<<(n_nodes + 255) / 256, 256, 0, stream>>>(
        (const float4*)acc, feat, w0, w1, out, n_nodes);
}